// MultiResImplicitFeature_38817914422140
// MI455X (gfx1250) — compile-verified
//
#include <hip/hip_runtime.h>

#define NFEAT 56
constexpr int BLOCK = 256;

typedef float v4f __attribute__((ext_vector_type(4)));  // native vector: OK for NT builtins

// ---------------------------------------------------------------------------
// Re-layout grid [8, R, R, R] (channel-major) -> [R, R, R, 8] (channels-last)
// so each trilerp corner is two contiguous 16B loads instead of 8 strided 4B.
// ---------------------------------------------------------------------------
__global__ __launch_bounds__(256) void transpose_grid(const float* __restrict__ g,
                                                      float* __restrict__ t, int R) {
    int v = blockIdx.x * blockDim.x + threadIdx.x;
    int nvox = R * R * R;
    if (v >= nvox) return;
    float vals[8];
#pragma unroll
    for (int c = 0; c < 8; ++c) vals[c] = g[(size_t)c * nvox + v];
    v4f* o = (v4f*)(t + (size_t)v * 8);
    v4f lo = {vals[0], vals[1], vals[2], vals[3]};
    v4f hi = {vals[4], vals[5], vals[6], vals[7]};
    o[0] = lo;  // transposed grids are the hot gather set: default (RT) stores,
    o[1] = hi;  // we WANT these resident in the 192MB L2
}

// ---------------------------------------------------------------------------
// 8-channel trilinear sample. USE_T: channels-last layout (b128 gathers),
// else original [C,R,R,R] layout (scalar gathers).
// Matches reference: i0 = clip(floor(c),0,R-1), i1 = clip(i0+1,0,R-1),
// w = c - floor(c) (unclamped).
// ---------------------------------------------------------------------------
template <bool USE_T>
__device__ __forceinline__ void trilerp8(const float* __restrict__ g, int R,
                                         float px, float py, float pz,
                                         float* __restrict__ acc) {
    float h = 0.5f * (float)(R - 1);
    float cx = (px + 1.0f) * h;
    float cy = (py + 1.0f) * h;
    float cz = (pz + 1.0f) * h;
    float fx = floorf(cx), fy = floorf(cy), fz = floorf(cz);
    float wx = cx - fx, wy = cy - fy, wz = cz - fz;
    int ix0 = min(max((int)fx, 0), R - 1);
    int iy0 = min(max((int)fy, 0), R - 1);
    int iz0 = min(max((int)fz, 0), R - 1);
    int ix1 = min(ix0 + 1, R - 1);
    int iy1 = min(iy0 + 1, R - 1);
    int iz1 = min(iz0 + 1, R - 1);
    float wxa[2] = {1.0f - wx, wx};
    float wya[2] = {1.0f - wy, wy};
    float wza[2] = {1.0f - wz, wz};
    int xs[2] = {ix0, ix1}, ys[2] = {iy0, iy1}, zs[2] = {iz0, iz1};
#pragma unroll
    for (int c = 0; c < 8; ++c) acc[c] = 0.0f;
#pragma unroll
    for (int dz = 0; dz < 2; ++dz)
#pragma unroll
        for (int dy = 0; dy < 2; ++dy)
#pragma unroll
            for (int dx = 0; dx < 2; ++dx) {
                float w = wza[dz] * wya[dy] * wxa[dx];
                int vox = (zs[dz] * R + ys[dy]) * R + xs[dx];
                if (USE_T) {
                    const v4f* p4 = (const v4f*)(g + (size_t)vox * 8);
                    v4f lo = p4[0];   // global_load_b128
                    v4f hi = p4[1];   // global_load_b128
                    acc[0] += w * lo.x; acc[1] += w * lo.y;
                    acc[2] += w * lo.z; acc[3] += w * lo.w;
                    acc[4] += w * hi.x; acc[5] += w * hi.y;
                    acc[6] += w * hi.z; acc[7] += w * hi.w;
                } else {
                    size_t cs = (size_t)R * R * R;
#pragma unroll
                    for (int c = 0; c < 8; ++c) acc[c] += w * g[(size_t)c * cs + vox];
                }
            }
}

// ---------------------------------------------------------------------------
// Main kernel: one thread per point; LDS-staged, NT, coalesced b128 stores.
// ---------------------------------------------------------------------------
template <bool USE_T>
__global__ __launch_bounds__(BLOCK) void featurize(
    const float* __restrict__ x,
    const float* __restrict__ g0, const float* __restrict__ g1,
    const float* __restrict__ g2, const float* __restrict__ g3,
    float* __restrict__ out, int n) {
    __shared__ float lds[BLOCK][57];  // 57-float rows: conflict-free scalar access

    const int tid = threadIdx.x;
    const int base_pt = blockIdx.x * BLOCK;
    const int pt = base_pt + tid;

    float px = 0.0f, py = 0.0f, pz = 0.0f;
    if (pt < n) {
        const float* xp = x + (size_t)3 * pt;
        px = __builtin_nontemporal_load(xp + 0);  // x streamed once: NT load
        py = __builtin_nontemporal_load(xp + 1);
        pz = __builtin_nontemporal_load(xp + 2);
    }

    float feat[NFEAT];

    // Positional encoding: feat[l*3+d] = sin((l+1)*pi/2*x_d), +12 for cos.
    const float HPI = 1.5707963267948966f;
    float p3[3] = {px, py, pz};
#pragma unroll
    for (int l = 0; l < 4; ++l)
#pragma unroll
        for (int d = 0; d < 3; ++d) {
            float a = (float)(l + 1) * HPI * p3[d];
            feat[l * 3 + d]      = __sinf(a);  // v_sin_f32
            feat[12 + l * 3 + d] = __cosf(a);  // v_cos_f32
        }

    trilerp8<USE_T>(g0, 16,  px, py, pz, feat + 24);
    trilerp8<USE_T>(g1, 32,  px, py, pz, feat + 32);
    trilerp8<USE_T>(g2, 64,  px, py, pz, feat + 40);
    trilerp8<USE_T>(g3, 128, px, py, pz, feat + 48);

    // Stage row in LDS (stride-57 rows -> conflict-free across lanes).
#pragma unroll
    for (int f = 0; f < NFEAT; ++f) lds[tid][f] = feat[f];
    __syncthreads();

    // Coalesced non-temporal b128 stores: wave writes contiguous 512B/instr,
    // NT hint keeps the 448MB output stream from evicting grids out of L2.
    const int npts_blk = min(BLOCK, n - base_pt);
    const int total4 = npts_blk * (NFEAT / 4);       // float4 count
    float* outb = out + (size_t)base_pt * NFEAT;     // 224B-aligned (14*16)
#pragma unroll 1
    for (int i4 = tid; i4 < total4; i4 += BLOCK) {
        int p = i4 / (NFEAT / 4);                    // point within block
        int f = (i4 - p * (NFEAT / 4)) * 4;          // feature (mult of 4, <56)
        v4f v = {lds[p][f], lds[p][f + 1], lds[p][f + 2], lds[p][f + 3]};
        __builtin_nontemporal_store(v, (v4f*)(outb + (size_t)i4 * 4));
    }
}

extern "C" void kernel_launch(void* const* d_in, const int* in_sizes, int n_in,
                              void* d_out, int out_size, void* d_ws, size_t ws_size,
                              hipStream_t stream) {
    const float* x  = (const float*)d_in[0];
    const float* g0 = (const float*)d_in[1];
    const float* g1 = (const float*)d_in[2];
    const float* g2 = (const float*)d_in[3];
    const float* g3 = (const float*)d_in[4];
    float* out = (float*)d_out;
    const int n = in_sizes[0] / 3;

    const size_t v0 = 16 * 16 * 16, v1 = 32 * 32 * 32, v2 = 64 * 64 * 64, v3 = 128 * 128 * 128;
    const size_t need = (size_t)8 * (v0 + v1 + v2 + v3) * sizeof(float);  // ~73.1 MB

    const int nblk = (n + BLOCK - 1) / BLOCK;

    if (d_ws != nullptr && ws_size >= need) {
        float* t0 = (float*)d_ws;
        float* t1 = t0 + 8 * v0;
        float* t2 = t1 + 8 * v1;
        float* t3 = t2 + 8 * v2;
        transpose_grid<<<(int)((v0 + 255) / 256), 256, 0, stream>>>(g0, t0, 16);
        transpose_grid<<<(int)((v1 + 255) / 256), 256, 0, stream>>>(g1, t1, 32);
        transpose_grid<<<(int)((v2 + 255) / 256), 256, 0, stream>>>(g2, t2, 64);
        transpose_grid<<<(int)((v3 + 255) / 256), 256, 0, stream>>>(g3, t3, 128);
        featurize<true><<<nblk, BLOCK, 0, stream>>>(x, t0, t1, t2, t3, out, n);
    } else {
        featurize<false><<<nblk, BLOCK, 0, stream>>>(x, g0, g1, g2, g3, out, n);
    }
}